// inter_prediction_80418967650321
// MI455X (gfx1250) — compile-verified
//
#include <hip/hip_runtime.h>

typedef __attribute__((ext_vector_type(2))) float vf2;
typedef __attribute__((ext_vector_type(8))) float vf8;

#define BLOCK 256
#define QPW 16                   // queries per wave (one 16-wide WMMA tile)
#define QPB ((BLOCK / 32) * QPW) // 128 queries per block
#define CHUNK_REFS 2048          // refs staged in LDS per pass (32 KB)
#define TILES (CHUNK_REFS / 16)
#define BIGD 3.0e37f
#define PADC 1.0e12f             // padded refs -> astronomically far
#define EPSV 1e-8f

// Branchless sorted top-3 (d0 <= d1 <= d2) insertion with index tracking.
__device__ __forceinline__ void top3_insert(float d, int i,
    float& d0, float& d1, float& d2, int& i0, int& i1, int& i2) {
  const bool lt0 = d < d0, lt1 = d < d1, lt2 = d < d2;
  d2 = lt1 ? d1 : (lt2 ? d : d2);
  i2 = lt1 ? i1 : (lt2 ? i : i2);
  d1 = lt0 ? d0 : (lt1 ? d : d1);
  i1 = lt0 ? i0 : (lt1 ? i : i1);
  d0 = lt0 ? d : d0;
  i0 = lt0 ? i : i0;
}

__global__ __launch_bounds__(BLOCK) void knn3_interp_wmma_kernel(
    const float* __restrict__ xyz1, const float* __restrict__ point1,
    const float* __restrict__ xyz2, const float* __restrict__ motion,
    float* __restrict__ out, int N1, int Q, int C) {
  __shared__ __align__(16) float lds[CHUNK_REFS * 4]; // (x,y,z,|r|^2) per ref

  const int tid   = threadIdx.x;
  const int lane  = tid & 31;
  const int half  = lane >> 4;   // 0: K0/K1 side, 1: K2/K3 side
  const int qlane = lane & 15;
  const int wave  = tid >> 5;

  const int qraw = blockIdx.x * QPB + wave * QPW + qlane;
  const int qn   = qraw < Q ? qraw : (Q - 1);   // clamp so full wave runs WMMA
  const int n2   = qn / C;
  const int c    = qn - n2 * C;

  // Query position = xyz2[n2] + motion[n2, c]  (motion is [N2,C,3] => qn*3)
  const float qx = xyz2[n2 * 3 + 0] + motion[qn * 3 + 0];
  const float qy = xyz2[n2 * 3 + 1] + motion[qn * 3 + 1];
  const float qz = xyz2[n2 * 3 + 2] + motion[qn * 3 + 2];

  // B matrix (4x16, loop invariant): col n = (-2qx, -2qy, -2qz, 1)
  vf2 B;
  B.x = half ? (-2.0f * qz) : (-2.0f * qx);
  B.y = half ? 1.0f         : (-2.0f * qy);

  float d0 = BIGD, d1 = BIGD, d2 = BIGD;
  int   i0 = 0,    i1 = 0,    i2 = 0;
  vf8 czero = {};

  for (int base = 0; base < N1; base += CHUNK_REFS) {
    // Cooperative staging of (x, y, z, |r|^2); pad tail with far-away points.
    for (int r = tid; r < CHUNK_REFS; r += BLOCK) {
      const int  g  = base + r;
      const bool ok = g < N1;
      const int  gi = ok ? g : 0;
      float x = xyz1[gi * 3 + 0], y = xyz1[gi * 3 + 1], z = xyz1[gi * 3 + 2];
      x = ok ? x : PADC; y = ok ? y : PADC; z = ok ? z : PADC;
      float4 v; v.x = x; v.y = y; v.z = z; v.w = x * x + y * y + z * z;
      ((float4*)lds)[r] = v;
    }
    __syncthreads();

    const float2* lds2 = (const float2*)lds;
    for (int t = 0; t < TILES; ++t) {
      // A matrix (16x4): lane<16 -> (x,y)=K0,K1 ; lane>=16 -> (z,|r|^2)=K2,K3
      const float2 av = lds2[(t * 16 + qlane) * 2 + half];
      vf2 A; A.x = av.x; A.y = av.y;
      // D[m][n] = -2*q_n . r_m + |r_m|^2  (dist minus per-query constant |q|^2)
      vf8 D = __builtin_amdgcn_wmma_f32_16x16x4_f32(
          false, A, false, B, (short)0, czero, false, false);

      // Cheap tile filter: min of this lane's 8 candidates vs current 3rd-best.
      // The expensive cndmask insertion block is skipped (s_cbranch_execz)
      // whenever no lane in the wave can improve its top-3 -- the common case
      // once the running top-3 stabilizes. WMMA stays outside the divergent
      // region, so EXEC is all-ones at every v_wmma.
      const float m = fminf(fminf(fminf(D[0], D[1]), fminf(D[2], D[3])),
                            fminf(fminf(D[4], D[5]), fminf(D[6], D[7])));
      if (__any(m < d2)) {
        const int ref0 = base + t * 16 + half * 8;
        #pragma unroll
        for (int v = 0; v < 8; ++v)
          top3_insert(D[v], ref0 + v, d0, d1, d2, i0, i1, i2);
      }
    }
    __syncthreads();
  }

  // Merge top-3 of lane n with lane n+16 (same query, other 8 rows per tile).
  float pd0 = __shfl_xor(d0, 16, 32);
  float pd1 = __shfl_xor(d1, 16, 32);
  float pd2 = __shfl_xor(d2, 16, 32);
  int   pi0 = __shfl_xor(i0, 16, 32);
  int   pi1 = __shfl_xor(i1, 16, 32);
  int   pi2 = __shfl_xor(i2, 16, 32);
  top3_insert(pd0, pi0, d0, d1, d2, i0, i1, i2);
  top3_insert(pd1, pi1, d0, d1, d2, i0, i1, i2);
  top3_insert(pd2, pi2, d0, d1, d2, i0, i1, i2);

  // Restore true distances (+|q|^2), clamp, inverse-distance weights.
  const float qsq = qx * qx + qy * qy + qz * qz;
  const float dd0 = fmaxf(d0 + qsq, 0.0f) + EPSV;
  const float dd1 = fmaxf(d1 + qsq, 0.0f) + EPSV;
  const float dd2 = fmaxf(d2 + qsq, 0.0f) + EPSV;
  const float w0 = 1.0f / dd0, w1 = 1.0f / dd1, w2 = 1.0f / dd2;
  const float inv = 1.0f / fmaxf(w0 + w1 + w2, 3.0f);
  const float res = (w0 * point1[i0 * C + c] +
                     w1 * point1[i1 * C + c] +
                     w2 * point1[i2 * C + c]) * inv;
  if (half == 0 && qraw < Q) out[qraw] = res;
}

extern "C" void kernel_launch(void* const* d_in, const int* in_sizes, int n_in,
                              void* d_out, int out_size, void* d_ws, size_t ws_size,
                              hipStream_t stream) {
  (void)n_in; (void)d_ws; (void)ws_size;
  const float* xyz1   = (const float*)d_in[0]; // [N1,3]
  const float* point1 = (const float*)d_in[1]; // [N1,C]
  const float* xyz2   = (const float*)d_in[2]; // [N2,3]
  const float* motion = (const float*)d_in[3]; // [N2,C,3]
  float* out = (float*)d_out;                  // [N2,C]

  const int N1 = in_sizes[0] / 3;
  const int C  = in_sizes[1] / N1;
  const int N2 = in_sizes[2] / 3;
  const int Q  = N2 * C;                       // == out_size

  const int blocks = (Q + QPB - 1) / QPB;
  knn3_interp_wmma_kernel<<<blocks, BLOCK, 0, stream>>>(
      xyz1, point1, xyz2, motion, out, N1, Q, C);
}